// B_Splines_85779086836036
// MI455X (gfx1250) — compile-verified
//
#include <hip/hip_runtime.h>

#define NPTS    8192
#define NCOEF   128
#define DEGREE  3
#define NSPANS  125           // interior spans; valid span index in [3, 127]
#define TILES_PER_WAVE 8      // each wave computes a 16 x 128 output strip

typedef __attribute__((ext_vector_type(2))) float v2f;
typedef __attribute__((ext_vector_type(8))) float v8f;

// ---------------------------------------------------------------------------
// Phase 1: evaluate both 1-D cubic splines (8192 x-points with coefs,
// 8192 t-points with coefs_2) into workspace.  Local de Boor basis_funs:
// only 4 basis functions are nonzero for a cubic spline, identical math to
// the reference's full Cox-de-Boor recursion.
// ---------------------------------------------------------------------------
__global__ __launch_bounds__(256) void bspline_1d_kernel(
    const float* __restrict__ x, const float* __restrict__ t,
    const float* __restrict__ knots,
    const float* __restrict__ coefs, const float* __restrict__ coefs2,
    float* __restrict__ sx, float* __restrict__ st, int n)
{
  int gid = blockIdx.x * blockDim.x + threadIdx.x;
  if (gid >= 2 * n) return;
  const bool second = gid >= n;
  const int  i = second ? gid - n : gid;
  const float p = second ? t[i] : x[i];
  const float* cf = second ? coefs2 : coefs;

  // Knot span: interior knots are linspace => spacing 1/NSPANS; correct by one
  // step against the actual knot array to match knots[s] <= p < knots[s+1].
  int span = 3 + (int)floorf(p * (float)NSPANS);
  span = min(max(span, 3), 3 + NSPANS - 1);
  if (p < knots[span])               span = max(span - 1, 3);
  else if (p >= knots[span + 1])     span = min(span + 1, 3 + NSPANS - 1);

  float left[DEGREE + 1], right[DEGREE + 1], N[DEGREE + 1];
  N[0] = 1.0f;
  #pragma unroll
  for (int j = 1; j <= DEGREE; ++j) {
    left[j]  = p - knots[span + 1 - j];
    right[j] = knots[span + j] - p;
    float saved = 0.0f;
    #pragma unroll
    for (int r = 0; r < j; ++r) {
      float temp = N[r] / (right[r + 1] + left[j - r]);
      N[r]  = saved + right[r + 1] * temp;
      saved = left[j - r] * temp;
    }
    N[j] = saved;
  }

  float s = 0.0f;
  #pragma unroll
  for (int k = 0; k <= DEGREE; ++k) s += cf[span - DEGREE + k] * N[k];

  float* out = second ? st : sx;
  out[i] = s;
}

// ---------------------------------------------------------------------------
// Phase 2: out[i][j] = sx[i] * st[j] as a rank-1 WMMA (V_WMMA_F32_16X16X4_F32).
// A (16x4 f32): VGPR0 = K=0 (lanes 0-15), K=2 (lanes 16-31); VGPR1 = K=1, K=3.
//   -> put sx in K=0 only; all other K columns zero.
// B (4x16 f32): replicate st into every (VGPR, half) so B's K=0 row holds
//   st[n] no matter which physical slot K=0 occupies; rows K>0 are harmless
//   because A's K>0 columns are zero.  D = sx (outer) st exactly in f32.
// Output is 256 MB (> 192 MB L2) streamed once -> non-temporal stores.
// EXEC is all-ones throughout (exact grid, no early exits) as WMMA requires.
// ---------------------------------------------------------------------------
__global__ __launch_bounds__(256) void outer_wmma_kernel(
    const float* __restrict__ sx, const float* __restrict__ st,
    float* __restrict__ out)
{
  const int lane    = threadIdx.x & 31;
  const int wave    = (blockIdx.x * blockDim.x + threadIdx.x) >> 5;
  const int tileRow = wave >> 6;       // 512 row tiles of 16
  const int colGrp  = wave & 63;       // 64 column groups of 128
  const int row0    = tileRow * 16;
  const int col0    = colGrp * 128;

  const float sxv = sx[row0 + (lane & 15)];
  v2f a;
  a.x = (lane < 16) ? sxv : 0.0f;      // K=0 column = spline_x, K=2 half = 0
  a.y = 0.0f;                          // K=1 / K=3 columns = 0

  const int halfOff = (lane >> 4) << 3;   // lanes 16..31 hold rows M+8
  const int nOff    = lane & 15;

  #pragma unroll
  for (int tIdx = 0; tIdx < TILES_PER_WAVE; ++tIdx) {
    const int c = col0 + tIdx * 16;
    const float stv = st[c + nOff];
    v2f b; b.x = stv; b.y = stv;       // st replicated across all K rows

    v8f acc = {};
    acc = __builtin_amdgcn_wmma_f32_16x16x4_f32(
        /*neg_a=*/false, a, /*neg_b=*/false, b,
        /*c_mod=*/(short)0, acc, /*reuse_a=*/false, /*reuse_b=*/false);

    #pragma unroll
    for (int v = 0; v < 8; ++v) {
      const size_t off = (size_t)(row0 + v + halfOff) * (size_t)NPTS
                       + (size_t)(c + nOff);
      __builtin_nontemporal_store(acc[v], out + off);
    }
  }
}

// ---------------------------------------------------------------------------
extern "C" void kernel_launch(void* const* d_in, const int* in_sizes, int n_in,
                              void* d_out, int out_size, void* d_ws, size_t ws_size,
                              hipStream_t stream)
{
  const float* x      = (const float*)d_in[0];
  const float* t      = (const float*)d_in[1];
  const float* knots  = (const float*)d_in[2];
  const float* coefs  = (const float*)d_in[3];
  const float* coefs2 = (const float*)d_in[4];
  float* out = (float*)d_out;

  float* sx = (float*)d_ws;          // 8192 floats
  float* st = sx + NPTS;             // 8192 floats (64 KB total, L2-resident)

  const int n = in_sizes[0];         // 8192

  bspline_1d_kernel<<<(2 * n + 255) / 256, 256, 0, stream>>>(
      x, t, knots, coefs, coefs2, sx, st, n);

  const int totalWaves   = (NPTS / 16) * (NPTS / (16 * TILES_PER_WAVE)); // 512*64
  const int totalThreads = totalWaves * 32;                              // 1,048,576
  outer_wmma_kernel<<<totalThreads / 256, 256, 0, stream>>>(sx, st, out);
}